// Attention_43465069035693
// MI455X (gfx1250) — compile-verified
//
#include <hip/hip_runtime.h>
#include <math.h>

// Problem constants (from reference)
constexpr int B    = 256;
constexpr int S    = 196;
constexpr int RNN  = 1024;
constexpr int ATTH = 512;

typedef __attribute__((ext_vector_type(2))) float v2f;
typedef __attribute__((ext_vector_type(8))) float v8f;

// ---------------------------------------------------------------------------
// Kernel 1: att_h[B][ATTH] = h[B][RNN] @ w_h2att[ATTH][RNN]^T + b_h2att
// One 16x16 tile per wave using V_WMMA_F32_16X16X4_F32 (fp32 WMMA, K-step 4).
//
// ISA layouts (CDNA5 ISA):
//   A 16x4 f32: lane%16 = M; VGPR v holds K = 2*(lane/16) + v
//   B 4x16 f32 (mirror): lane%16 = N; VGPR v holds K = 2*(lane/16) + v
//     B[k][n] = w_h2att[n][k]  =>  both A and B stream contiguously along K.
//   C/D 16x16 f32: lane%16 = N; VGPR v holds M = 8*(lane/16) + v
// ---------------------------------------------------------------------------
__global__ __launch_bounds__(256)
void h2att_wmma_kernel(const float* __restrict__ h,
                       const float* __restrict__ w,      // [ATTH][RNN]
                       const float* __restrict__ bias,   // [ATTH]
                       float* __restrict__ att_h) {      // [B][ATTH]
  const int wave = threadIdx.x >> 5;                 // 0..7
  const int lane = threadIdx.x & 31;
  const int tile = blockIdx.x * 8 + wave;            // 0..511
  constexpr int TILES_N = ATTH / 16;                 // 32
  const int tm = tile / TILES_N;                     // 0..15
  const int tn = tile % TILES_N;                     // 0..31

  const int l16   = lane & 15;
  const int khalf = (lane >> 4) << 1;                // 0 or 2

  const float* __restrict__ arow = h + (size_t)(tm * 16 + l16) * RNN + khalf;
  const float* __restrict__ brow = w + (size_t)(tn * 16 + l16) * RNN + khalf;

  v8f acc = {};
#pragma unroll 8
  for (int k = 0; k < RNN; k += 4) {
    v2f a = *(const v2f*)(arow + k);
    v2f b = *(const v2f*)(brow + k);
    acc = __builtin_amdgcn_wmma_f32_16x16x4_f32(
        /*neg_a=*/false, a, /*neg_b=*/false, b,
        /*c_mod=*/(short)0, acc, /*reuse_a=*/false, /*reuse_b=*/false);
  }

  const float bn    = bias[tn * 16 + l16];
  const int   mbase = tm * 16 + ((lane >> 4) << 3);  // +0 or +8
#pragma unroll
  for (int v = 0; v < 8; ++v) {
    att_h[(size_t)(mbase + v) * ATTH + tn * 16 + l16] = acc[v] + bn;
  }
}

// ---------------------------------------------------------------------------
// Kernel 2: scores[b][s] = sum_a tanh(p_att[b,s,a] + att_h[b,a]) * w_alpha[a]
//           + b_alpha
// One wave32 per (b,s) row; float4 loads, shuffle reduction.
// ---------------------------------------------------------------------------
__global__ __launch_bounds__(256)
void score_kernel(const float* __restrict__ p_att,    // [B][S][ATTH]
                  const float* __restrict__ att_h,    // [B][ATTH]
                  const float* __restrict__ w_alpha,  // [ATTH]
                  const float* __restrict__ b_alpha,  // [1]
                  float* __restrict__ scores) {       // [B][S]
  const int wave = threadIdx.x >> 5;
  const int lane = threadIdx.x & 31;
  const int row  = blockIdx.x * 8 + wave;             // < B*S (divides exactly)
  const int b    = row / S;

  const float* __restrict__ p  = p_att + (size_t)row * ATTH;
  const float* __restrict__ ah = att_h + (size_t)b * ATTH;

  float partial = 0.0f;
#pragma unroll
  for (int j = 0; j < 4; ++j) {
    const int c = (lane + j * 32) * 4;                // covers 512 floats
    float4 pv = *(const float4*)(p + c);
    float4 av = *(const float4*)(ah + c);
    float4 wv = *(const float4*)(w_alpha + c);
    partial += tanhf(pv.x + av.x) * wv.x;
    partial += tanhf(pv.y + av.y) * wv.y;
    partial += tanhf(pv.z + av.z) * wv.z;
    partial += tanhf(pv.w + av.w) * wv.w;
  }
#pragma unroll
  for (int off = 16; off > 0; off >>= 1)
    partial += __shfl_down(partial, off, 32);
  if (lane == 0) scores[row] = partial + b_alpha[0];
}

// ---------------------------------------------------------------------------
// Kernel 3: softmax over S per batch + att_res[b][d] = sum_s w[b,s]*att_feats
// One block (256 threads) per batch; each thread owns 4 columns of RNN=1024.
// att_feats (205 MB) is streamed exactly once.
// ---------------------------------------------------------------------------
__global__ __launch_bounds__(256)
void attres_kernel(const float* __restrict__ scores,    // [B][S]
                   const float* __restrict__ att_feats, // [B][S][RNN]
                   float* __restrict__ out) {           // [B][RNN]
  __shared__ float red[256];
  __shared__ float sw[256];

  const int b = blockIdx.x;
  const int t = threadIdx.x;

  const float sc = (t < S) ? scores[b * S + t] : -INFINITY;

  // max-reduce
  red[t] = sc;
  __syncthreads();
#pragma unroll
  for (int off = 128; off > 0; off >>= 1) {
    if (t < off) red[t] = fmaxf(red[t], red[t + off]);
    __syncthreads();
  }
  const float m = red[0];
  __syncthreads();

  // sum-reduce of exp
  const float e = (t < S) ? __expf(sc - m) : 0.0f;
  red[t] = e;
  __syncthreads();
#pragma unroll
  for (int off = 128; off > 0; off >>= 1) {
    if (t < off) red[t] += red[t + off];
    __syncthreads();
  }
  const float inv = 1.0f / red[0];
  sw[t] = e * inv;
  __syncthreads();

  // weighted sum: thread t owns columns [4t, 4t+3]
  const float* __restrict__ af = att_feats + (size_t)b * S * RNN + t * 4;
  float4 acc = {0.f, 0.f, 0.f, 0.f};
  for (int s = 0; s < S; ++s) {
    if (s + 8 < S) __builtin_prefetch(af + (size_t)(s + 8) * RNN, 0, 0);
    const float  wgt = sw[s];
    const float4 v   = *(const float4*)(af + (size_t)s * RNN);
    acc.x += wgt * v.x;
    acc.y += wgt * v.y;
    acc.z += wgt * v.z;
    acc.w += wgt * v.w;
  }
  *(float4*)(out + (size_t)b * RNN + t * 4) = acc;
}

// ---------------------------------------------------------------------------
// Launch
// ---------------------------------------------------------------------------
extern "C" void kernel_launch(void* const* d_in, const int* in_sizes, int n_in,
                              void* d_out, int out_size, void* d_ws, size_t ws_size,
                              hipStream_t stream) {
  const float* h        = (const float*)d_in[0];
  const float* att_feats= (const float*)d_in[1];
  const float* p_att    = (const float*)d_in[2];
  const float* w_h2att  = (const float*)d_in[3];
  const float* b_h2att  = (const float*)d_in[4];
  const float* w_alpha  = (const float*)d_in[5];
  const float* b_alpha  = (const float*)d_in[6];
  float* out = (float*)d_out;

  float* att_h  = (float*)d_ws;                 // B*ATTH floats  (512 KB)
  float* scores = att_h + (size_t)B * ATTH;     // B*S floats     (196 KB)

  // K1: 512 tiles / 8 waves per block = 64 blocks
  h2att_wmma_kernel<<<64, 256, 0, stream>>>(h, w_h2att, b_h2att, att_h);
  // K2: B*S = 50176 rows / 8 waves per block = 6272 blocks
  score_kernel<<<(B * S) / 8, 256, 0, stream>>>(p_att, att_h, w_alpha, b_alpha, scores);
  // K3: one block per batch
  attres_kernel<<<B, 256, 0, stream>>>(scores, att_feats, out);
}